// GraphTracker_12610023981445
// MI455X (gfx1250) — compile-verified
//
#include <hip/hip_runtime.h>
#include <hip/hip_bf16.h>
#include <math.h>

typedef __attribute__((ext_vector_type(2))) float v2f;
typedef __attribute__((ext_vector_type(8))) float v8f;

// Problem constants (match reference setup_inputs)
constexpr int Dd  = 512;   // feature dim
constexpr int Msz = 512;   // tra rows
constexpr int Nsz = 512;   // det rows
constexpr int NP  = 513;   // augmented dim

// GEMM tiling
constexpr int TM = 128, TN = 128, KC = 16;
constexpr int LDSS = 20;   // padded LDS row stride (floats): conflict-free & 16B aligned

// ---------------------------------------------------------------------------
// gfx1250 async global->LDS copy (ASYNCcnt-tracked, bypasses VGPRs).
// ldsOff = LDS byte offset (low 32 bits of a generic pointer to __shared__).
// ---------------------------------------------------------------------------
__device__ __forceinline__ void async_copy_b128(uint32_t ldsOff, const float* g)
{
    asm volatile("global_load_async_to_lds_b128 %0, %1, off"
                 :: "v"(ldsOff), "v"((uint64_t)(uintptr_t)g)
                 : "memory");
}
__device__ __forceinline__ uint32_t lds_off(const void* p)
{
    return (uint32_t)(uintptr_t)p;   // generic LDS pointer: addr[31:0] == LDS offset
}

// ---------------------------------------------------------------------------
// Kernel 1: inverse row norms of tra (first G*M entries) and det (next G*N).
// One wave per 512-float row; float4 loads; wave32 shuffle reduce.
// ---------------------------------------------------------------------------
__global__ __launch_bounds__(256) void norms_kernel(const float* __restrict__ det,
                                                    const float* __restrict__ tra,
                                                    float* __restrict__ invn, int G)
{
    const int lane = threadIdx.x & 31;
    const int row  = blockIdx.x * (blockDim.x >> 5) + (threadIdx.x >> 5);
    const int total = G * Msz * 2;
    if (row >= total) return;
    const float* src = (row < G * Msz)
        ? (tra + (size_t)row * Dd)
        : (det + (size_t)(row - G * Msz) * Dd);
    float s = 0.f;
    #pragma unroll
    for (int d = lane * 4; d < Dd; d += 128) {
        float4 v = *(const float4*)(src + d);
        s += v.x * v.x + v.y * v.y + v.z * v.z + v.w * v.w;
    }
    #pragma unroll
    for (int off = 16; off; off >>= 1) s += __shfl_xor(s, off);
    if (lane == 0) invn[row] = 1.0f / sqrtf(s);
}

// ---------------------------------------------------------------------------
// Kernel 2: per-graph GEMM corr = tra @ det^T via v_wmma_f32_16x16x4_f32,
// fused with cosine normalization and K = -affinity/lambda, written into the
// [G,513,513] output buffer (the Sinkhorn kernel matrix).
// Block = 256 threads = 8 waves; block tile 128x128; wave tile 32x64 (2x4 WMMA).
// Double-buffered LDS stages filled by async global->LDS b128 copies:
// stage i+1 streams in (ASYNCcnt) while stage i feeds the WMMA pipe.
// ---------------------------------------------------------------------------
__global__ __launch_bounds__(256) void gemm_affinity_kernel(
    const float* __restrict__ det, const float* __restrict__ tra,
    const float* __restrict__ eps, const float* __restrict__ invn,
    float* __restrict__ K, int G)
{
    __shared__ float As[2][TM][LDSS];
    __shared__ float Bs[2][TN][LDSS];

    const int g  = blockIdx.z;
    const int n0 = blockIdx.x * TN;
    const int m0 = blockIdx.y * TM;
    const int tid  = threadIdx.x;
    const int lane = tid & 31;
    const int w    = tid >> 5;
    const int wm   = (w & 3) * 32;   // wave M offset (4 waves along M)
    const int wn   = (w >> 2) * 64;  // wave N offset (2 waves along N)

    const float negInvLambd = -1.0f / (__expf(eps[0]) + 0.03f);
    const float* Ag  = tra  + (size_t)g * Msz * Dd;
    const float* Bg  = det  + (size_t)g * Nsz * Dd;
    const float* in1 = invn + (size_t)g * Msz;                    // tra inv-norms
    const float* in2 = invn + (size_t)G * Msz + (size_t)g * Nsz;  // det inv-norms

    // Per-thread stage-copy assignment: rows r0 and r0+64, one float4 each.
    const int r0 = tid >> 2;            // 0..63
    const int c4 = (tid & 3) * 4;       // 0/4/8/12
    const float* gA0 = Ag + (size_t)(m0 + r0)      * Dd + c4;
    const float* gA1 = Ag + (size_t)(m0 + r0 + 64) * Dd + c4;
    const float* gB0 = Bg + (size_t)(n0 + r0)      * Dd + c4;
    const float* gB1 = Bg + (size_t)(n0 + r0 + 64) * Dd + c4;
    uint32_t lA0[2], lA1[2], lB0[2], lB1[2];
    #pragma unroll
    for (int b = 0; b < 2; ++b) {
        lA0[b] = lds_off(&As[b][r0][c4]);
        lA1[b] = lds_off(&As[b][r0 + 64][c4]);
        lB0[b] = lds_off(&Bs[b][r0][c4]);
        lB1[b] = lds_off(&Bs[b][r0 + 64][c4]);
    }

    v8f acc[2][4] = {};

    // Prologue: stream stage 0 into buffer 0 (4 async ops per wave).
    async_copy_b128(lA0[0], gA0);
    async_copy_b128(lA1[0], gA1);
    async_copy_b128(lB0[0], gB0);
    async_copy_b128(lB1[0], gB1);

    int buf = 0;
    for (int k0 = 0; k0 < Dd; k0 += KC) {
        __syncthreads();   // all waves done reading buf^1 before we overwrite it
        if (k0 + KC < Dd) {
            const int nb = buf ^ 1;
            const int ko = k0 + KC;
            async_copy_b128(lA0[nb], gA0 + ko);
            async_copy_b128(lA1[nb], gA1 + ko);
            async_copy_b128(lB0[nb], gB0 + ko);
            async_copy_b128(lB1[nb], gB1 + ko);
            // in-order completion: <=4 outstanding => current stage landed
            asm volatile("s_wait_asynccnt 0x4" ::: "memory");
        } else {
            asm volatile("s_wait_asynccnt 0x0" ::: "memory");
        }
        __syncthreads();   // current stage visible to all waves

        const int lrow = lane & 15;
        #pragma unroll
        for (int kk = 0; kk < KC; kk += 4) {
            const int krow = kk + 2 * (lane >> 4);   // ISA 16x4 f32 A layout
            v2f a0 = *(const v2f*)&As[buf][wm + lrow][krow];
            v2f a1 = *(const v2f*)&As[buf][wm + 16 + lrow][krow];
            v2f b0 = *(const v2f*)&Bs[buf][wn + lrow][krow];
            v2f b1 = *(const v2f*)&Bs[buf][wn + 16 + lrow][krow];
            v2f b2 = *(const v2f*)&Bs[buf][wn + 32 + lrow][krow];
            v2f b3 = *(const v2f*)&Bs[buf][wn + 48 + lrow][krow];
            acc[0][0] = __builtin_amdgcn_wmma_f32_16x16x4_f32(false, a0, false, b0, (short)0, acc[0][0], false, false);
            acc[0][1] = __builtin_amdgcn_wmma_f32_16x16x4_f32(false, a0, false, b1, (short)0, acc[0][1], false, false);
            acc[0][2] = __builtin_amdgcn_wmma_f32_16x16x4_f32(false, a0, false, b2, (short)0, acc[0][2], false, false);
            acc[0][3] = __builtin_amdgcn_wmma_f32_16x16x4_f32(false, a0, false, b3, (short)0, acc[0][3], false, false);
            acc[1][0] = __builtin_amdgcn_wmma_f32_16x16x4_f32(false, a1, false, b0, (short)0, acc[1][0], false, false);
            acc[1][1] = __builtin_amdgcn_wmma_f32_16x16x4_f32(false, a1, false, b1, (short)0, acc[1][1], false, false);
            acc[1][2] = __builtin_amdgcn_wmma_f32_16x16x4_f32(false, a1, false, b2, (short)0, acc[1][2], false, false);
            acc[1][3] = __builtin_amdgcn_wmma_f32_16x16x4_f32(false, a1, false, b3, (short)0, acc[1][3], false, false);
        }
        buf ^= 1;
    }

    // Epilogue: D layout (16x16 f32): VGPR r -> M=r (lanes 0-15) / M=r+8 (16-31)
    const int half = lane >> 4;
    const int cn   = lane & 15;
    float* Kg = K + (size_t)g * NP * NP;
    #pragma unroll
    for (int im = 0; im < 2; ++im) {
        #pragma unroll
        for (int in = 0; in < 4; ++in) {
            #pragma unroll
            for (int r = 0; r < 8; ++r) {
                const int m = m0 + wm + im * 16 + r + 8 * half;
                const int n = n0 + wn + in * 16 + cn;
                Kg[(size_t)m * NP + n] = acc[im][in][r] * in1[m] * in2[n] * negInvLambd;
            }
        }
    }
}

// ---------------------------------------------------------------------------
// Kernel 3: fill augmented last row/col with K = -alpha/lambda.
// ---------------------------------------------------------------------------
__global__ void fill_bins_kernel(const float* __restrict__ alpha,
                                 const float* __restrict__ eps,
                                 float* __restrict__ K)
{
    const float bval = -alpha[0] / (__expf(eps[0]) + 0.03f);
    float* Kg = K + (size_t)blockIdx.x * NP * NP;
    for (int j = threadIdx.x; j < NP; j += blockDim.x) {
        Kg[(size_t)(NP - 1) * NP + j] = bval;   // last row
        Kg[(size_t)j * NP + (NP - 1)] = bval;   // last col
    }
}

// ---------------------------------------------------------------------------
// Kernel 4: Sinkhorn via potentials. One workgroup (512 threads = 16 waves)
// per graph. u,v live in LDS; K stays in global (L2-resident; 134.7MB < 192MB).
//   col pass: thread-per-column (coalesced down rows)
//   row pass: wave-per-row, lane-strided + shuffle LSE reduction (coalesced)
// Final pass overwrites K in place with exp(K + u_i + v_j).
// ---------------------------------------------------------------------------
__global__ __launch_bounds__(512) void sinkhorn_kernel(float* __restrict__ K)
{
    __shared__ float u_s[NP];
    __shared__ float v_s[NP];
    float* Kg = K + (size_t)blockIdx.x * NP * NP;
    const int tid  = threadIdx.x;
    const int lane = tid & 31;
    const int wid  = tid >> 5;
    const int nw   = blockDim.x >> 5;

    const float log_norm  = -__logf((float)(Msz + Nsz));  // log(1/1024)
    const float log_extra = __logf((float)Msz);           // bins get norm*512

    for (int i = tid; i < NP; i += blockDim.x) u_s[i] = 0.f;
    __syncthreads();

    for (int it = 0; it < 8; ++it) {
        // v_j = log_b[j] - LSE_m(K[m,j] + u[m])   (online LSE)
        for (int j = tid; j < NP; j += blockDim.x) {
            float mx = -INFINITY, s = 0.f;
            for (int m = 0; m < NP; ++m) {
                float x  = Kg[(size_t)m * NP + j] + u_s[m];
                float nm = fmaxf(mx, x);
                s  = s * __expf(mx - nm) + __expf(x - nm);
                mx = nm;
            }
            float logb = log_norm + ((j == NP - 1) ? log_extra : 0.f);
            v_s[j] = logb - (mx + __logf(s));
        }
        __syncthreads();
        // u_i = log_a[i] - LSE_n(K[i,n] + v[n])
        for (int i = wid; i < NP; i += nw) {
            float mx = -INFINITY, s = 0.f;
            for (int n = lane; n < NP; n += 32) {
                float x  = Kg[(size_t)i * NP + n] + v_s[n];
                float nm = fmaxf(mx, x);
                s  = s * __expf(mx - nm) + __expf(x - nm);
                mx = nm;
            }
            #pragma unroll
            for (int off = 16; off; off >>= 1) {
                float m2 = __shfl_xor(mx, off);
                float s2 = __shfl_xor(s, off);
                float nm = fmaxf(mx, m2);
                s  = s * __expf(mx - nm) + s2 * __expf(m2 - nm);
                mx = nm;
            }
            if (lane == 0) {
                float loga = log_norm + ((i == NP - 1) ? log_extra : 0.f);
                u_s[i] = loga - (mx + __logf(s));
            }
        }
        __syncthreads();
    }

    // out = exp(K + u_i + v_j), in place (single HBM-writing pass)
    for (int i = wid; i < NP; i += nw) {
        const float ui = u_s[i];
        for (int j = lane; j < NP; j += 32) {
            const size_t idx = (size_t)i * NP + j;
            Kg[idx] = __expf(Kg[idx] + ui + v_s[j]);
        }
    }
}

// ---------------------------------------------------------------------------
extern "C" void kernel_launch(void* const* d_in, const int* in_sizes, int n_in,
                              void* d_out, int out_size, void* d_ws, size_t ws_size,
                              hipStream_t stream) {
    const float* det   = (const float*)d_in[0];   // [G, N, D]
    const float* tra   = (const float*)d_in[1];   // [G, M, D]
    const float* alpha = (const float*)d_in[2];   // [1]
    const float* eps   = (const float*)d_in[3];   // [1]
    float* out  = (float*)d_out;                  // [G, 513, 513]
    float* invn = (float*)d_ws;                   // G*(M+N) floats

    const int G = in_sizes[0] / (Nsz * Dd);

    // 1) inverse norms: one wave per row, 8 waves per block
    const int rows = G * Msz * 2;
    norms_kernel<<<(rows + 7) / 8, 256, 0, stream>>>(det, tra, invn, G);

    // 2) WMMA GEMM + cosine norm + (-1/lambda) scale -> K in d_out
    dim3 gg(Nsz / TN, Msz / TM, G);
    gemm_affinity_kernel<<<gg, 256, 0, stream>>>(det, tra, eps, invn, out, G);

    // 3) augmented bins row/col
    fill_bins_kernel<<<G, 256, 0, stream>>>(alpha, eps, out);

    // 4) Sinkhorn (potentials) + final exp, in place on d_out
    sinkhorn_kernel<<<G, 512, 0, stream>>>(out);
}